// MyModelBlock_74105365725492
// MI455X (gfx1250) — compile-verified
//
#include <hip/hip_runtime.h>
#include <math.h>

typedef float v2f __attribute__((ext_vector_type(2)));
typedef float v8f __attribute__((ext_vector_type(8)));

#define DH 128

__device__ __forceinline__ void atomicMaxF(float* addr, float val) {
  // monotonic int-punning trick for float max (init must be -inf)
  if (val >= 0.0f) atomicMax((int*)addr, __float_as_int(val));
  else             atomicMin((unsigned int*)addr, (unsigned int)__float_as_int(val));
}

__global__ void fill_kernel(float* __restrict__ p, float v, long long n) {
  long long i = (long long)blockIdx.x * blockDim.x + threadIdx.x;
  if (i < n) p[i] = v;
}

// -------- Z = H @ W^T via V_WMMA_F32_16X16X4_F32 --------
// H: [nrows,128] row-major, W: [128,128] row-major (Z[m,n] = sum_k H[m,k]*W[n,k])
// Block = 256 threads = 8 waves; block handles one 16-row stripe; wave w -> cols [16w,16w+16).
__global__ __launch_bounds__(256) void gemm_w_kernel(
    const float* __restrict__ H, const float* __restrict__ W,
    float* __restrict__ Z, const int* nrows_p, int nrows_cap) {
  int nrows = nrows_cap;
  if (nrows_p) { int v = *nrows_p; if (v < nrows) nrows = v; }
  const int m0 = blockIdx.x * 16;
  if (m0 >= nrows) return;                       // block-uniform -> EXEC stays full
  const int lane = threadIdx.x & 31;
  const int n0   = (threadIdx.x >> 5) * 16;
  const int l15  = lane & 15;
  const int kh   = (lane >> 4) << 1;             // 0 (lanes 0-15) or 2 (lanes 16-31)
  int arow = m0 + l15; if (arow >= nrows) arow = nrows - 1;   // clamped read, store guarded
  const float* hr = H + (size_t)arow * DH + kh;
  const float* wr = W + (size_t)(n0 + l15) * DH + kh;
  v8f acc = {};
#pragma unroll
  for (int k0 = 0; k0 < DH; k0 += 4) {
    const float2 av = *(const float2*)(hr + k0);
    const float2 bv = *(const float2*)(wr + k0);
    v2f a; a.x = av.x; a.y = av.y;               // A 16x4: K pair per half-wave
    v2f b; b.x = bv.x; b.y = bv.y;               // B 4x16: N=lane&15, same K split
    acc = __builtin_amdgcn_wmma_f32_16x16x4_f32(false, a, false, b, (short)0, acc,
                                                false, false);
  }
  const int srow = m0 + ((lane >> 4) << 3);      // C/D: lanes16-31 hold M+8
  const int scol = n0 + l15;
#pragma unroll
  for (int r = 0; r < 8; ++r) {
    int rr = srow + r;
    if (rr < nrows) Z[(size_t)rr * DH + scol] = acc[r];
  }
}

// -------- OUT = relu(H @ U^T + ZN) (fused zi GEMM + finalize) --------
__global__ __launch_bounds__(256) void gemm_u_relu_kernel(
    const float* __restrict__ H, const float* __restrict__ U,
    const float* __restrict__ ZN, float* __restrict__ OUT,
    const int* nrows_p, int nrows_cap) {
  int nrows = nrows_cap;
  if (nrows_p) { int v = *nrows_p; if (v < nrows) nrows = v; }
  const int m0 = blockIdx.x * 16;
  if (m0 >= nrows) return;
  const int lane = threadIdx.x & 31;
  const int n0   = (threadIdx.x >> 5) * 16;
  const int l15  = lane & 15;
  const int kh   = (lane >> 4) << 1;
  int arow = m0 + l15; if (arow >= nrows) arow = nrows - 1;
  const float* hr = H + (size_t)arow * DH + kh;
  const float* ur = U + (size_t)(n0 + l15) * DH + kh;
  v8f acc = {};
#pragma unroll
  for (int k0 = 0; k0 < DH; k0 += 4) {
    const float2 av = *(const float2*)(hr + k0);
    const float2 bv = *(const float2*)(ur + k0);
    v2f a; a.x = av.x; a.y = av.y;
    v2f b; b.x = bv.x; b.y = bv.y;
    acc = __builtin_amdgcn_wmma_f32_16x16x4_f32(false, a, false, b, (short)0, acc,
                                                false, false);
  }
  const int srow = m0 + ((lane >> 4) << 3);
  const int scol = n0 + l15;
#pragma unroll
  for (int r = 0; r < 8; ++r) {
    int rr = srow + r;
    if (rr < nrows) {
      float v = acc[r] + ZN[(size_t)rr * DH + scol];
      OUT[(size_t)rr * DH + scol] = v > 0.0f ? v : 0.0f;
    }
  }
}

// -------- per-node: S1 = Z . a[0:128], S2 = Z . a[128:256] (wave per node) --------
__global__ __launch_bounds__(256) void rowdot_kernel(
    const float* __restrict__ Z, const float* __restrict__ a,
    float* __restrict__ S1, float* __restrict__ S2,
    const int* nrows_p, int nrows_cap) {
  int nrows = nrows_cap;
  if (nrows_p) { int v = *nrows_p; if (v < nrows) nrows = v; }
  const int node = blockIdx.x * 8 + (threadIdx.x >> 5);   // wave-uniform
  if (node >= nrows) return;
  const int lane = threadIdx.x & 31;
  float4 zv = ((const float4*)(Z + (size_t)node * DH))[lane];
  float4 av = ((const float4*)a)[lane];
  float4 bv = ((const float4*)(a + DH))[lane];
  float p1 = zv.x * av.x + zv.y * av.y + zv.z * av.z + zv.w * av.w;
  float p2 = zv.x * bv.x + zv.y * bv.y + zv.z * bv.z + zv.w * bv.w;
#pragma unroll
  for (int off = 16; off > 0; off >>= 1) {
    p1 += __shfl_xor(p1, off, 32);
    p2 += __shfl_xor(p2, off, 32);
  }
  if (lane == 0) { S1[node] = p1; S2[node] = p2; }
}

// -------- per-edge logits + segment max --------
__global__ void edge_logits_kernel(
    const float* __restrict__ S1, const float* __restrict__ S2,
    const float* __restrict__ d, const int* __restrict__ src,
    const int* __restrict__ dst, const float* __restrict__ V,
    const float* __restrict__ a, float* __restrict__ E, float* __restrict__ M,
    int ne) {
  int i = blockIdx.x * blockDim.x + threadIdx.x;
  if (i >= ne) return;
  float e = S1[src[i]] + S2[dst[i]] + d[i] * V[0] * a[2 * DH];
  e = e > 0.0f ? e : 0.01f * e;                  // leaky_relu (slope 0.01)
  E[i] = e;
  atomicMaxF(&M[dst[i]], e);
}

// -------- per-edge exp + segment sum --------
__global__ void edge_exp_kernel(float* __restrict__ E, const float* __restrict__ M,
                                const int* __restrict__ dst, float* __restrict__ SS,
                                int ne) {
  int i = blockIdx.x * blockDim.x + threadIdx.x;
  if (i >= ne) return;
  int dd = dst[i];
  float v = expf(E[i] - M[dd]);
  E[i] = v;
  atomicAdd(&SS[dd], v);
}

// -------- aggregation: ZN[dst] += (EX/SS[dst]) * Z[src]  (wave per edge) --------
__global__ __launch_bounds__(256) void aggregate_kernel(
    const float* __restrict__ EX, const float* __restrict__ SS,
    const float* __restrict__ Z, const int* __restrict__ src,
    const int* __restrict__ dst, float* __restrict__ ZN, int ne) {
  const int e = blockIdx.x * 8 + (threadIdx.x >> 5);      // wave-uniform
  if (e >= ne) return;
  const int lane = threadIdx.x & 31;
  const int s = src[e], dd = dst[e];
  const float w = EX[e] / SS[dd];
  float4 zv = ((const float4*)(Z + (size_t)s * DH))[lane];
  float* o = ZN + (size_t)dd * DH + lane * 4;
  atomicAdd(o + 0, w * zv.x);
  atomicAdd(o + 1, w * zv.y);
  atomicAdd(o + 2, w * zv.z);
  atomicAdd(o + 3, w * zv.w);
}

static inline int ceil_div(long long a, long long b) { return (int)((a + b - 1) / b); }

extern "C" void kernel_launch(void* const* d_in, const int* in_sizes, int n_in,
                              void* d_out, int out_size, void* d_ws, size_t ws_size,
                              hipStream_t stream) {
  const float* attr = (const float*)d_in[0];
  const float* d0   = (const float*)d_in[1];
  const float* d1   = (const float*)d_in[2];
  const int*   src0 = (const int*)d_in[3];
  const int*   dst0 = (const int*)d_in[4];
  const int*   src1 = (const int*)d_in[5];
  const int*   dst1 = (const int*)d_in[6];
  const int*   nd0p = (const int*)d_in[7];   // n_dst0 (device scalar)
  const int*   nd1p = (const int*)d_in[8];   // n_dst1 (device scalar)
  const float* W0   = (const float*)d_in[9];
  const float* U0   = (const float*)d_in[10];
  const float* V0   = (const float*)d_in[11];
  const float* a0   = (const float*)d_in[12];
  const float* W1   = (const float*)d_in[13];
  const float* U1   = (const float*)d_in[14];
  const float* V1   = (const float*)d_in[15];
  const float* a1   = (const float*)d_in[16];

  const int N0  = in_sizes[0] / DH;          // 100000
  const int E0  = in_sizes[1];               // 640000
  const int E1  = in_sizes[2];               // 320000
  const int ND1 = out_size / DH;             // 20000 (host-known from output size)
  const int Emax = E0 > E1 ? E0 : E1;

  // workspace layout (floats)
  float* p  = (float*)d_ws;
  float* Z  = p;  p += (size_t)N0 * DH;      // z projection (both layers)
  float* ZN = p;  p += (size_t)N0 * DH;      // layer-0 zn, overwritten in-place -> H1
  float* ZN1 = p; p += (size_t)ND1 * DH;     // layer-1 zn
  float* S1 = p;  p += N0;
  float* S2 = p;  p += N0;
  float* EX = p;  p += Emax;
  float* M  = p;  p += N0;
  float* SS = p;  p += N0;

  const float NEG_INF = -__builtin_huge_valf();

  // ---------------- Layer 0 ----------------
  fill_kernel<<<ceil_div((long long)N0 * DH, 256), 256, 0, stream>>>(ZN, 0.0f, (long long)N0 * DH);
  fill_kernel<<<ceil_div(N0, 256), 256, 0, stream>>>(M, NEG_INF, N0);
  fill_kernel<<<ceil_div(N0, 256), 256, 0, stream>>>(SS, 0.0f, N0);

  gemm_w_kernel<<<ceil_div(N0, 16), 256, 0, stream>>>(attr, W0, Z, nullptr, N0);
  rowdot_kernel<<<ceil_div(N0, 8), 256, 0, stream>>>(Z, a0, S1, S2, nullptr, N0);
  edge_logits_kernel<<<ceil_div(E0, 256), 256, 0, stream>>>(S1, S2, d0, src0, dst0, V0, a0, EX, M, E0);
  edge_exp_kernel<<<ceil_div(E0, 256), 256, 0, stream>>>(EX, M, dst0, SS, E0);
  aggregate_kernel<<<ceil_div(E0, 8), 256, 0, stream>>>(EX, SS, Z, src0, dst0, ZN, E0);
  // H1 = relu(attr @ U0^T + ZN), written in-place over ZN (elementwise-safe)
  gemm_u_relu_kernel<<<ceil_div(N0, 16), 256, 0, stream>>>(attr, U0, ZN, ZN, nd0p, N0);

  // ---------------- Layer 1 (h = ZN holds H1 rows [0, n_dst0)) ----------------
  fill_kernel<<<ceil_div((long long)ND1 * DH, 256), 256, 0, stream>>>(ZN1, 0.0f, (long long)ND1 * DH);
  fill_kernel<<<ceil_div(N0, 256), 256, 0, stream>>>(M, NEG_INF, N0);
  fill_kernel<<<ceil_div(N0, 256), 256, 0, stream>>>(SS, 0.0f, N0);

  gemm_w_kernel<<<ceil_div(N0, 16), 256, 0, stream>>>(ZN, W1, Z, nd0p, N0);
  rowdot_kernel<<<ceil_div(N0, 8), 256, 0, stream>>>(Z, a1, S1, S2, nd0p, N0);
  edge_logits_kernel<<<ceil_div(E1, 256), 256, 0, stream>>>(S1, S2, d1, src1, dst1, V1, a1, EX, M, E1);
  edge_exp_kernel<<<ceil_div(E1, 256), 256, 0, stream>>>(EX, M, dst1, SS, E1);
  aggregate_kernel<<<ceil_div(E1, 8), 256, 0, stream>>>(EX, SS, Z, src1, dst1, ZN1, E1);
  gemm_u_relu_kernel<<<ceil_div(ND1, 16), 256, 0, stream>>>(ZN, U1, ZN1, (float*)d_out, nd1p, ND1);
}